// adLMLIFSNN_11235634446821
// MI455X (gfx1250) — compile-verified
//
#include <hip/hip_runtime.h>

typedef __attribute__((ext_vector_type(2))) float v2f;
typedef __attribute__((ext_vector_type(8))) float v8f;

#define LDK 20  // padded LDS row (16 K + 4): 80B row stride keeps 16B alignment
                // for async B128 LDS writes; 20*m mod 64 -> 16 distinct banks.

// ---------------------------------------------------------------------------
// Zero-padded weight copy: W[N x K] -> Wp[Npad x K]
// ---------------------------------------------------------------------------
__global__ void pad_w_kernel(const float* __restrict__ W, float* __restrict__ Wp,
                             int N, int K, int Npad) {
    int idx = blockIdx.x * blockDim.x + threadIdx.x;
    int total = Npad * K;
    if (idx >= total) return;
    int n = idx / K;
    int k = idx - n * K;
    Wp[idx] = (n < N) ? W[n * K + k] : 0.0f;
}

// ---------------------------------------------------------------------------
// Async global -> LDS direct copy (16B per lane), tracked by ASYNCcnt.
// ---------------------------------------------------------------------------
__device__ __forceinline__ void async_copy_b128(unsigned lds_off, const float* g) {
    asm volatile("global_load_async_to_lds_b128 %0, %1, off"
                 :: "v"(lds_off), "v"(g)
                 : "memory");
}

__device__ __forceinline__ void wait_async0() {
    asm volatile("s_wait_asynccnt 0x0" ::: "memory");
}

__device__ __forceinline__ unsigned lds_off32(const float* p) {
    // Flat->LDS aperture mapping truncates to addr[31:0] (wave-relative offset).
    return (unsigned)(unsigned long long)(const void*)p;
}

// ---------------------------------------------------------------------------
// One K-block (16) of micro-k WMMAs: each wave computes a 32x32 output from
// its 2 M-subtiles x 2 N-subtiles (4 f32 16x16 accumulators).
// ---------------------------------------------------------------------------
__device__ __forceinline__ void microk_block(const float* __restrict__ As,
                                             const float* __restrict__ Bs,
                                             v8f (&acc)[2][2],
                                             int wm, int wn, int l15, int khi2) {
    #pragma unroll
    for (int kk = 0; kk < 16; kk += 4) {
        const int ks = kk + khi2;
        // A fragments (16x4): lane -> row m = l15; VGPR0/1 -> K = ks/ks+1
        v2f a0, a1;
        a0.x = As[(wm * 32 + l15) * LDK + ks];
        a0.y = As[(wm * 32 + l15) * LDK + ks + 1];
        a1.x = As[(wm * 32 + 16 + l15) * LDK + ks];
        a1.y = As[(wm * 32 + 16 + l15) * LDK + ks + 1];
        // B fragments (4x16): lane -> column n = l15; VGPR0/1 -> K = ks/ks+1
        v2f b0, b1;
        b0.x = Bs[(wn * 32 + l15) * LDK + ks];
        b0.y = Bs[(wn * 32 + l15) * LDK + ks + 1];
        b1.x = Bs[(wn * 32 + 16 + l15) * LDK + ks];
        b1.y = Bs[(wn * 32 + 16 + l15) * LDK + ks + 1];

        acc[0][0] = __builtin_amdgcn_wmma_f32_16x16x4_f32(
            false, a0, false, b0, (short)0, acc[0][0], false, false);
        acc[0][1] = __builtin_amdgcn_wmma_f32_16x16x4_f32(
            false, a0, false, b1, (short)0, acc[0][1], false, false);
        acc[1][0] = __builtin_amdgcn_wmma_f32_16x16x4_f32(
            false, a1, false, b0, (short)0, acc[1][0], false, false);
        acc[1][1] = __builtin_amdgcn_wmma_f32_16x16x4_f32(
            false, a1, false, b1, (short)0, acc[1][1], false, false);
    }
}

// ---------------------------------------------------------------------------
// Z[M x Npad] = X[M x K] * Wp[Npad x K]^T   (fp32 WMMA 16x16x4)
// Block: 256 threads = 8 waves, block tile 128(M) x 64(N).
// Waves arranged 4(M) x 2(N); each wave computes 32x32 (2x2 accumulators).
// Double-buffered LDS filled by GLOBAL_LOAD_ASYNC_TO_LDS_B128; one barrier
// per K-block; next tile's fill overlaps the current WMMA burst.
// ---------------------------------------------------------------------------
__global__ __launch_bounds__(256)
void gemm_nt_wmma_f32(const float* __restrict__ X, const float* __restrict__ Wp,
                      float* __restrict__ Z, int M, int Npad, int K, int ldz) {
    __shared__ float As[2][128 * LDK];
    __shared__ float Bs[2][64 * LDK];

    const int tx   = threadIdx.x;
    const int lane = tx & 31;
    const int wave = tx >> 5;     // 0..7
    const int wm   = wave >> 1;   // 4 wave rows (M)
    const int wn   = wave & 1;    // 2 wave cols (N)
    const int bM   = blockIdx.y * 128;
    const int bN   = blockIdx.x * 64;

    const int l15  = lane & 15;
    const int khi2 = (lane >> 4) << 1;   // 0 for lanes 0-15, 2 for lanes 16-31

    v8f acc[2][2] = {};

    const int lrow = tx >> 2;          // 0..63
    const int lq   = (tx & 3) * 4;     // 0,4,8,12

    const float* srcA0 = X  + (size_t)(bM + lrow)      * K + lq;   // rows 0..63
    const float* srcA1 = X  + (size_t)(bM + lrow + 64) * K + lq;   // rows 64..127
    const float* srcB  = Wp + (size_t)(bN + lrow)      * K + lq;   // rows 0..63

    unsigned dA0[2], dA1[2], dB[2];
    #pragma unroll
    for (int p = 0; p < 2; ++p) {
        dA0[p] = lds_off32(&As[p][lrow * LDK + lq]);
        dA1[p] = lds_off32(&As[p][(lrow + 64) * LDK + lq]);
        dB [p] = lds_off32(&Bs[p][lrow * LDK + lq]);
    }

    const int Kmain = K & ~15;
    const int nb    = Kmain >> 4;

    if (nb > 0) {   // prime buffer 0
        async_copy_b128(dA0[0], srcA0);
        async_copy_b128(dA1[0], srcA1);
        async_copy_b128(dB [0], srcB);
    }

    for (int k = 0; k < nb; ++k) {
        const int cur = k & 1;
        const int nxt = cur ^ 1;
        wait_async0();       // this wave's copies into buf[cur] are in LDS
        __syncthreads();     // all waves' data visible; all done reading buf[nxt]
        if (k + 1 < nb) {
            const int kb = (k + 1) << 4;
            async_copy_b128(dA0[nxt], srcA0 + kb);
            async_copy_b128(dA1[nxt], srcA1 + kb);
            async_copy_b128(dB [nxt], srcB  + kb);
        }
        microk_block(&As[cur][0], &Bs[cur][0], acc, wm, wn, l15, khi2);
    }

    // ---- K remainder (e.g. 700 = 43*16 + 12), executed once ----
    if (Kmain < K) {
        __syncthreads();     // everyone finished reading the last buffer
        const int kg = Kmain + lq;
        float* eA0 = &As[0][lrow * LDK + lq];
        float* eA1 = &As[0][(lrow + 64) * LDK + lq];
        float* eB  = &Bs[0][lrow * LDK + lq];
        #pragma unroll
        for (int j = 0; j < 4; ++j) {
            eA0[j] = (kg + j < K) ? srcA0[Kmain + j] : 0.0f;
            eA1[j] = (kg + j < K) ? srcA1[Kmain + j] : 0.0f;
            eB [j] = (kg + j < K) ? srcB [Kmain + j] : 0.0f;
        }
        __syncthreads();
        microk_block(&As[0][0], &Bs[0][0], acc, wm, wn, l15, khi2);
    }

    // ---- store C/D (VGPR r: lanes 0-15 -> row r, lanes 16-31 -> row r+8) ----
    const int mWave = bM + wm * 32 + ((lane >> 4) << 3);
    const int nWave = bN + wn * 32 + l15;
    #pragma unroll
    for (int mt = 0; mt < 2; ++mt) {
        #pragma unroll
        for (int nt = 0; nt < 2; ++nt) {
            const int mBase = mWave + mt * 16;
            const int nCol  = nWave + nt * 16;
            #pragma unroll
            for (int r = 0; r < 8; ++r) {
                Z[(size_t)(mBase + r) * ldz + nCol] = acc[mt][nt][r];
            }
        }
    }
}

// ---------------------------------------------------------------------------
// BatchNorm stage 1: per-(chunk, channel) partial sum / sumsq over rows.
// ---------------------------------------------------------------------------
__global__ void bn_partial_kernel(const float* __restrict__ Z, int ldz, int N,
                                  int rowsPerChunk, int M,
                                  float* __restrict__ psum, float* __restrict__ psq) {
    int i = blockIdx.x * blockDim.x + threadIdx.x;   // channel
    int c = blockIdx.y;                              // chunk
    if (i >= N) return;
    int r0 = c * rowsPerChunk;
    int r1 = r0 + rowsPerChunk;
    if (r1 > M) r1 = M;
    float s = 0.0f, q = 0.0f;
    for (int r = r0; r < r1; ++r) {
        float v = Z[(size_t)r * ldz + i];
        s += v;
        q += v * v;
    }
    psum[(size_t)c * N + i] = s;
    psq [(size_t)c * N + i] = q;
}

// ---------------------------------------------------------------------------
// BatchNorm stage 2: reduce chunks, fold into scale/shift.
// ---------------------------------------------------------------------------
__global__ void bn_finalize_kernel(const float* __restrict__ psum,
                                   const float* __restrict__ psq,
                                   int N, int C, float invM,
                                   const float* __restrict__ g,
                                   const float* __restrict__ bb,
                                   float* __restrict__ scale,
                                   float* __restrict__ shift) {
    int i = blockIdx.x * blockDim.x + threadIdx.x;
    if (i >= N) return;
    float s = 0.0f, q = 0.0f;
    for (int c = 0; c < C; ++c) {
        s += psum[(size_t)c * N + i];
        q += psq [(size_t)c * N + i];
    }
    float mu  = s * invM;
    float var = q * invM - mu * mu;
    float sc  = g[i] * rsqrtf(var + 1e-5f);
    scale[i] = sc;
    shift[i] = bb[i] - mu * sc;
}

// ---------------------------------------------------------------------------
// LIF recurrence: one thread per (batch, channel); loop over T.
//   Wc = alpha*Wc + a*U + b*S
//   U  = beta*(U - S) + (1-beta)*(z_t - Wc)     (THETA = 1)
//   S  = heaviside(U - 1)
// ---------------------------------------------------------------------------
__global__ void lif_seq_kernel(const float* __restrict__ Z, int ldz,
                               const float* __restrict__ scale,
                               const float* __restrict__ shift,
                               const float* __restrict__ alpha_p,
                               const float* __restrict__ beta_p,
                               const float* __restrict__ a_p,
                               const float* __restrict__ b_p,
                               float* __restrict__ out,
                               int Bsz, int T, int N) {
    int id = blockIdx.x * blockDim.x + threadIdx.x;
    if (id >= Bsz * N) return;
    int b = id / N;
    int i = id - b * N;

    const float ALPHA_LO = 0.8187307530779818f;   // exp(-1/5)
    const float ALPHA_HI = 0.9607894391523232f;   // exp(-1/25)
    const float BETA_LO  = 0.9672161004820059f;   // exp(-1/30)
    const float BETA_HI  = 0.9917012918570596f;   // exp(-1/120)

    float alpha = fminf(fmaxf(alpha_p[i], ALPHA_LO), ALPHA_HI);
    float beta  = fminf(fmaxf(beta_p[i],  BETA_LO),  BETA_HI);
    float pa    = fminf(fmaxf(a_p[i], -1.0f), 1.0f);
    float pb    = fminf(fmaxf(b_p[i],  0.0f), 2.0f);
    float sc    = scale[i];
    float sh    = shift[i];
    float omb   = 1.0f - beta;

    float U = 0.0f, Wc = 0.0f, S = 0.0f;
    const float* zrow = Z   + (size_t)b * T * ldz + i;
    float*       orow = out + (size_t)b * T * N   + i;

    for (int t = 0; t < T; ++t) {
        float zt = zrow[(size_t)t * ldz] * sc + sh;
        Wc = alpha * Wc + pa * U + pb * S;
        U  = beta * (U - S) + omb * (zt - Wc);
        S  = (U > 1.0f) ? 1.0f : 0.0f;
        orow[(size_t)t * N] = S;
    }
}

// ---------------------------------------------------------------------------
extern "C" void kernel_launch(void* const* d_in, const int* in_sizes, int n_in,
                              void* d_out, int out_size, void* d_ws, size_t ws_size,
                              hipStream_t stream) {
    const int B = 64, T = 500, J = 700, H = 512;
    const int M = B * T;                // 32000
    const int NPAD_MAX = 704;           // 700 rounded up to 64
    const int CHUNK = 256;              // rows per BN chunk
    const int NCHUNK = M / CHUNK;       // 125

    const float* x = (const float*)d_in[0];

    // Workspace carve-up (floats)
    float* Zbuf  = (float*)d_ws;                           // M * 704
    float* actA  = Zbuf  + (size_t)M * NPAD_MAX;           // M * 512
    float* Wpad  = actA  + (size_t)M * H;                  // 704 * 704
    float* psum  = Wpad  + (size_t)NPAD_MAX * NPAD_MAX;    // 125 * 704
    float* psq   = psum  + (size_t)NCHUNK * NPAD_MAX;      // 125 * 704
    float* scale = psq   + (size_t)NCHUNK * NPAD_MAX;      // 704
    float* shift = scale + NPAD_MAX;                       // 704

    // Layer descriptors: dims (out N, in K), input, output.
    // Layer-2 activations are parked in d_out: they are fully consumed by the
    // layer-3 GEMM before the final LIF kernel overwrites d_out (stream order).
    struct Layer { int N, K, Npad; const float* X; float* O; };
    Layer layers[3] = {
        { H, J, H,        x,              actA           },
        { H, H, H,        actA,           (float*)d_out  },
        { J, H, NPAD_MAX, (float*)d_out,  (float*)d_out  },
    };

    for (int l = 0; l < 3; ++l) {
        const Layer& L = layers[l];
        const float* W    = (const float*)d_in[1 + 7 * l + 0];
        const float* bn_g = (const float*)d_in[1 + 7 * l + 1];
        const float* bn_b = (const float*)d_in[1 + 7 * l + 2];
        const float* al   = (const float*)d_in[1 + 7 * l + 3];
        const float* be   = (const float*)d_in[1 + 7 * l + 4];
        const float* pa   = (const float*)d_in[1 + 7 * l + 5];
        const float* pb   = (const float*)d_in[1 + 7 * l + 6];

        // 1) zero-padded weights
        {
            int tot = L.Npad * L.K;
            pad_w_kernel<<<(tot + 255) / 256, 256, 0, stream>>>(W, Wpad, L.N, L.K, L.Npad);
        }
        // 2) GEMM: Zbuf[M x Npad] = X @ Wpad^T
        {
            dim3 grid(L.Npad / 64, M / 128);
            gemm_nt_wmma_f32<<<grid, 256, 0, stream>>>(L.X, Wpad, Zbuf, M, L.Npad, L.K, L.Npad);
        }
        // 3) BatchNorm stats -> scale/shift
        {
            dim3 grid1((L.N + 255) / 256, NCHUNK);
            bn_partial_kernel<<<grid1, 256, 0, stream>>>(Zbuf, L.Npad, L.N, CHUNK, M, psum, psq);
            bn_finalize_kernel<<<(L.N + 255) / 256, 256, 0, stream>>>(
                psum, psq, L.N, NCHUNK, 1.0f / (float)M, bn_g, bn_b, scale, shift);
        }
        // 4) LIF recurrence over T
        {
            int threads = B * L.N;
            lif_seq_kernel<<<(threads + 255) / 256, 256, 0, stream>>>(
                Zbuf, L.Npad, scale, shift, al, be, pa, pb, L.O, B, T, L.N);
        }
    }
}